// IntegratedColorReasoningModel_17463337025792
// MI455X (gfx1250) — compile-verified
//
#include <hip/hip_runtime.h>
#include <hip/hip_bf16.h>

typedef float v2f __attribute__((ext_vector_type(2)));
typedef float v8f __attribute__((ext_vector_type(8)));

#define HW 4096  // 64*64 pixels per channel

// ---------------------------------------------------------------------------
// Kernel A: mean over H*W for each (b, c) slab. One wave (32 lanes) per slab.
// Each lane reads 32 float4's (512B coalesced per wave-step), wave-reduces.
// This kernel is the entire runtime (201 MB @ 23.3 TB/s ~ 9 us).
// ---------------------------------------------------------------------------
__global__ void __launch_bounds__(256)
mean_rgb_kernel(const float* __restrict__ x, float* __restrict__ mean_rgb, int nslab) {
  const int wv   = threadIdx.x >> 5;
  const int lane = threadIdx.x & 31;
  const int slab = blockIdx.x * (blockDim.x >> 5) + wv;
  if (slab >= nslab) return;
  const float4* p = (const float4*)(x + (size_t)slab * HW);
  float s = 0.0f;
#pragma unroll
  for (int i = 0; i < 32; ++i) {
    float4 v = p[i * 32 + lane];
    s += (v.x + v.y) + (v.z + v.w);
  }
#pragma unroll
  for (int off = 16; off > 0; off >>= 1)
    s += __shfl_xor(s, off, 32);
  if (lane == 0) mean_rgb[slab] = s * (1.0f / (float)HW);
}

// ---------------------------------------------------------------------------
// Kernel B: classifier layers 1+2 with V_WMMA_F32_16X16X4_F32.
//   h  = relu(mean_rgb(16x3, K padded to 4) @ w1(3x128) + b1)
//   cs = relu(h(16x128) @ w2(128x64) + b2)
// One wave per 16-row batch tile; 4 waves (128 threads) per block.
// Fragment layouts per CDNA5 ISA 7.12.2 (f32, wave32):
//   A 16x4 : lanes 0-15 -> {K0,K1}, lanes 16-31 -> {K2,K3}, M = lane%16
//   B 4x16 : lanes 0-15 -> {K0,K1}, lanes 16-31 -> {K2,K3}, N = lane%16
//   C 16x16: VGPR r -> M = r + 8*(lane/16), N = lane%16
// ---------------------------------------------------------------------------
__global__ void __launch_bounds__(128)
mlp12_wmma_kernel(const float* __restrict__ mean_rgb,
                  const float* __restrict__ w1, const float* __restrict__ b1,
                  const float* __restrict__ w2, const float* __restrict__ b2,
                  float* __restrict__ cs) {
  __shared__ float h[4][16][132];  // 132 pad: row stride = 4 mod 64 banks

  const int wv   = threadIdx.x >> 5;
  const int lane = threadIdx.x & 31;
  const int lm   = lane & 15;
  const int half = lane >> 4;           // 0: K=0..1, 1: K=2..3
  const int r0   = (blockIdx.x * 4 + wv) * 16;

  // --- Layer 1: (16x3 pad->16x4) @ (3x128 pad->4x128) ---
  v2f a1;
  {
    const int row = r0 + lm;
    a1.x = mean_rgb[row * 3 + 2 * half];               // K = 0 or 2
    a1.y = half ? 0.0f : mean_rgb[row * 3 + 1];        // K = 1 or 3(=0 pad)
  }
#pragma unroll
  for (int c = 0; c < 8; ++c) {
    const int n = c * 16 + lm;
    v2f b;
    b.x = w1[(2 * half) * 128 + n];                    // K = 0 or 2
    b.y = half ? 0.0f : w1[128 + n];                   // K = 1 or 3(=0 pad)
    const float bias = b1[n];
    v8f acc;
#pragma unroll
    for (int r = 0; r < 8; ++r) acc[r] = bias;
    acc = __builtin_amdgcn_wmma_f32_16x16x4_f32(false, a1, false, b,
                                                (short)0, acc, false, false);
#pragma unroll
    for (int r = 0; r < 8; ++r)
      h[wv][r + 8 * half][n] = fmaxf(acc[r], 0.0f);    // relu, stage in LDS
  }
  __syncthreads();  // cross-lane LDS RAW within each wave's tile

  // --- Layer 2: (16x128) @ (128x64), K in 32 chained WMMA steps ---
#pragma unroll
  for (int c = 0; c < 4; ++c) {
    const int n = c * 16 + lm;
    const float bias = b2[n];
    v8f acc;
#pragma unroll
    for (int r = 0; r < 8; ++r) acc[r] = bias;
    for (int s = 0; s < 32; ++s) {
      const int k0 = 4 * s + 2 * half;
      v2f a, b;
      a.x = h[wv][lm][k0];
      a.y = h[wv][lm][k0 + 1];
      b.x = w2[k0 * 64 + n];
      b.y = w2[(k0 + 1) * 64 + n];
      acc = __builtin_amdgcn_wmma_f32_16x16x4_f32(false, a, false, b,
                                                  (short)0, acc, false, false);
    }
#pragma unroll
    for (int r = 0; r < 8; ++r)
      cs[(size_t)(r0 + r + 8 * half) * 64 + n] = fmaxf(acc[r], 0.0f);
  }
}

// ---------------------------------------------------------------------------
// Kernel C: per-row tail. One thread per batch row.
// ---------------------------------------------------------------------------
__device__ inline void softmax5(const float* l, float* p) {
  float m = l[0];
#pragma unroll
  for (int j = 1; j < 5; ++j) m = fmaxf(m, l[j]);
  float s = 0.0f;
#pragma unroll
  for (int j = 0; j < 5; ++j) { p[j] = expf(l[j] - m); s += p[j]; }
  const float inv = 1.0f / s;
#pragma unroll
  for (int j = 0; j < 5; ++j) p[j] *= inv;
}

__global__ void __launch_bounds__(256)
tail_kernel(const float* __restrict__ mean_rgb, const float* __restrict__ cs,
            const float* __restrict__ w3, const float* __restrict__ b3,
            const float* __restrict__ rel_w1, const float* __restrict__ rel_b1,
            const float* __restrict__ rel_w2, const float* __restrict__ rel_b2,
            const float* __restrict__ int_w1, const float* __restrict__ int_b1,
            const float* __restrict__ int_w2, const float* __restrict__ int_b2,
            const float* __restrict__ int_w3, const float* __restrict__ int_b3,
            const float* __restrict__ cw_p, const float* __restrict__ rw_p,
            float* __restrict__ out, int B) {
  const int i = blockIdx.x * blockDim.x + threadIdx.x;
  if (i >= B) return;

  const float r  = mean_rgb[i * 3 + 0];
  const float g  = mean_rgb[i * 3 + 1];
  const float bl = mean_rgb[i * 3 + 2];

  // classifier logits: cs(64) @ w3(64x5) + b3
  float cl[5];
#pragma unroll
  for (int j = 0; j < 5; ++j) cl[j] = b3[j];
  for (int k = 0; k < 64; ++k) {
    const float v = cs[(size_t)i * 64 + k];
#pragma unroll
    for (int j = 0; j < 5; ++j) cl[j] += v * w3[k * 5 + j];
  }
  float cp[5];
  softmax5(cl, cp);

  // color bin (first-hit priority, argmax fallback with first-index tie rule)
  int idx;
  if      (r > 0.5f && g < 0.3f && bl < 0.3f) idx = 0;
  else if (r > 0.5f && g > 0.3f && bl < 0.3f) idx = 1;
  else if (r > 0.5f && g > 0.5f && bl < 0.3f) idx = 2;
  else if (r < 0.3f && g > 0.5f && bl < 0.3f) idx = 3;
  else if (r < 0.3f && g < 0.3f && bl > 0.5f) idx = 4;
  else if (r > 0.5f && g < 0.3f && bl > 0.5f) idx = 5;
  else if (r > 0.7f && g > 0.7f && bl > 0.7f) idx = 6;
  else if (r < 0.3f && g < 0.3f && bl < 0.3f) idx = 7;
  else idx = (r >= g && r >= bl) ? 0 : ((g >= bl) ? 1 : 2);

  // relation MLP: one_hot(idx,8) @ rel_w1 selects row idx
  float rl[5];
#pragma unroll
  for (int j = 0; j < 5; ++j) rl[j] = rel_b2[j];
  for (int k = 0; k < 32; ++k) {
    const float hv = fmaxf(rel_w1[idx * 32 + k] + rel_b1[k], 0.0f);
#pragma unroll
    for (int j = 0; j < 5; ++j) rl[j] += hv * rel_w2[k * 5 + j];
  }
  float rp[5];
  softmax5(rl, rp);

  // integration MLP: comb(10) -> 64 -> 32 -> 5 (stream the 64-wide layer)
  float comb[10];
#pragma unroll
  for (int j = 0; j < 5; ++j) { comb[j] = cp[j]; comb[5 + j] = rp[j]; }
  float ih2[32];
#pragma unroll
  for (int t = 0; t < 32; ++t) ih2[t] = int_b2[t];
  for (int j = 0; j < 64; ++j) {
    float a = int_b1[j];
#pragma unroll
    for (int t = 0; t < 10; ++t) a += comb[t] * int_w1[t * 64 + j];
    a = fmaxf(a, 0.0f);
#pragma unroll
    for (int t = 0; t < 32; ++t) ih2[t] += a * int_w2[j * 32 + t];
  }
  float il[5];
#pragma unroll
  for (int j = 0; j < 5; ++j) il[j] = int_b3[j];
  for (int k = 0; k < 32; ++k) {
    const float v = fmaxf(ih2[k], 0.0f);
#pragma unroll
    for (int j = 0; j < 5; ++j) il[j] += v * int_w3[k * 5 + j];
  }
  float ip[5];
  softmax5(il, ip);

  // weighted final logits
  const float cw = cw_p[0], rw = rw_p[0];
  float fl[5], fp[5];
#pragma unroll
  for (int j = 0; j < 5; ++j) fl[j] = cw * cl[j] + rw * rl[j];
  softmax5(fl, fp);

  float* o = out + (size_t)i * 20;
#pragma unroll
  for (int j = 0; j < 5; ++j) {
    o[j]      = cp[j];
    o[5 + j]  = rp[j];
    o[10 + j] = ip[j];
    o[15 + j] = fp[j];
  }
}

// ---------------------------------------------------------------------------
extern "C" void kernel_launch(void* const* d_in, const int* in_sizes, int n_in,
                              void* d_out, int out_size, void* d_ws, size_t ws_size,
                              hipStream_t stream) {
  const float* x      = (const float*)d_in[0];
  const float* clf_w1 = (const float*)d_in[1];
  const float* clf_b1 = (const float*)d_in[2];
  const float* clf_w2 = (const float*)d_in[3];
  const float* clf_b2 = (const float*)d_in[4];
  const float* clf_w3 = (const float*)d_in[5];
  const float* clf_b3 = (const float*)d_in[6];
  const float* rel_w1 = (const float*)d_in[7];
  const float* rel_b1 = (const float*)d_in[8];
  const float* rel_w2 = (const float*)d_in[9];
  const float* rel_b2 = (const float*)d_in[10];
  const float* int_w1 = (const float*)d_in[11];
  const float* int_b1 = (const float*)d_in[12];
  const float* int_w2 = (const float*)d_in[13];
  const float* int_b2 = (const float*)d_in[14];
  const float* int_w3 = (const float*)d_in[15];
  const float* int_b3 = (const float*)d_in[16];
  const float* cw     = (const float*)d_in[17];
  const float* rw     = (const float*)d_in[18];
  float* out = (float*)d_out;

  const int B = in_sizes[0] / (3 * HW);      // 4096

  float* mean_rgb = (float*)d_ws;            // B*3 floats
  float* cs       = mean_rgb + (size_t)B * 3;  // B*64 floats

  const int nslab = B * 3;
  mean_rgb_kernel<<<(nslab + 7) / 8, 256, 0, stream>>>(x, mean_rgb, nslab);

  mlp12_wmma_kernel<<<B / 64, 128, 0, stream>>>(mean_rgb, clf_w1, clf_b1,
                                                clf_w2, clf_b2, cs);

  tail_kernel<<<(B + 255) / 256, 256, 0, stream>>>(
      mean_rgb, cs, clf_w3, clf_b3, rel_w1, rel_b1, rel_w2, rel_b2,
      int_w1, int_b1, int_w2, int_b2, int_w3, int_b3, cw, rw, out, B);
}